// DominantModel_17824114279158
// MI455X (gfx1250) — compile-verified
//
#include <hip/hip_runtime.h>

typedef __attribute__((ext_vector_type(2))) float v2f;
typedef __attribute__((ext_vector_type(8))) float v8f;

#define NNODES 12288
#define FDIM   512
#define HDIM   64

// ---------------------------------------------------------------------------
// Generic f32 WMMA GEMM: C[Nrows x Fout] = A[Nrows x K] * B[K x Fout]
// One wave computes one 16x16 tile of C using V_WMMA_F32_16X16X4_F32.
// A-frag (16x4 f32): lanes 0-15 -> M=lane, K={k,k+1}; lanes 16-31 -> K={k+2,k+3}
// B-frag (4x16 f32): lanes 0-15 -> K={k,k+1}, N=lane; lanes 16-31 -> K={k+2,k+3}
// C/D (16x16 f32):  VGPR j: lanes 0-15 M=j, lanes 16-31 M=j+8; N=lane&15
// ---------------------------------------------------------------------------
__global__ void gemm_wmma_f32(const float* __restrict__ A,
                              const float* __restrict__ B,
                              float* __restrict__ C,
                              int K, int Fout, int totalTiles, int tilesX) {
    const int lane = threadIdx.x & 31;
    const int wave = threadIdx.x >> 5;
    const int tile = blockIdx.x * (blockDim.x >> 5) + wave;
    if (tile >= totalTiles) return;   // wave-uniform: EXEC stays all-1s

    const int ty   = tile / tilesX;
    const int tx   = tile - ty * tilesX;
    const int row0 = ty * 16;
    const int col0 = tx * 16;

    const int mrow = lane & 15;
    const int koff = (lane >= 16) ? 2 : 0;

    const float* __restrict__ arow = A + (size_t)(row0 + mrow) * K;
    v8f acc = {};

#pragma unroll 4
    for (int k = 0; k < K; k += 4) {
        v2f a, b;
        a.x = arow[k + koff];
        a.y = arow[k + koff + 1];
        b.x = B[(size_t)(k + koff)     * Fout + col0 + mrow];
        b.y = B[(size_t)(k + koff + 1) * Fout + col0 + mrow];
        acc = __builtin_amdgcn_wmma_f32_16x16x4_f32(
            /*neg_a=*/false, a, /*neg_b=*/false, b,
            /*c_mod=*/(short)0, acc, /*reuse_a=*/false, /*reuse_b=*/false);
    }

    const int hi8 = (lane >> 4) << 3;           // +8 rows for lanes 16-31
#pragma unroll
    for (int j = 0; j < 8; ++j)
        C[(size_t)(row0 + hi8 + j) * Fout + col0 + mrow] = acc[j];
}

// ---------------------------------------------------------------------------
// struct = s @ s.T  (N x N output, K = HDIM = 64).  s is 3 MB -> L2 resident.
// Output is 604 MB streaming -> nontemporal stores to avoid L2 pollution.
// ---------------------------------------------------------------------------
__global__ void struct_wmma(const float* __restrict__ s,
                            float* __restrict__ out) {
    const int lane = threadIdx.x & 31;
    const int wave = threadIdx.x >> 5;
    const int tx   = blockIdx.x * 8 + wave;     // column tile 0..767
    const int ty   = blockIdx.y;                // row    tile 0..767
    const int row0 = ty * 16;
    const int col0 = tx * 16;

    const int mrow = lane & 15;
    const int koff = (lane >= 16) ? 2 : 0;

    const float* __restrict__ arow = s + (size_t)(row0 + mrow) * HDIM; // A row
    const float* __restrict__ brow = s + (size_t)(col0 + mrow) * HDIM; // B=s.T col

    v8f acc = {};
#pragma unroll
    for (int k = 0; k < HDIM; k += 4) {
        v2f a, b;
        a.x = arow[k + koff];
        a.y = arow[k + koff + 1];
        b.x = brow[k + koff];
        b.y = brow[k + koff + 1];
        acc = __builtin_amdgcn_wmma_f32_16x16x4_f32(
            false, a, false, b, (short)0, acc, false, false);
    }

    const int hi8 = (lane >> 4) << 3;
    const size_t base = (size_t)(row0 + hi8) * NNODES + col0 + mrow;
#pragma unroll
    for (int j = 0; j < 8; ++j)
        __builtin_nontemporal_store(acc[j], out + base + (size_t)j * NNODES);
}

// ---------------------------------------------------------------------------
// agg[n][f] = bias[f]
// ---------------------------------------------------------------------------
__global__ void bias_init(float* __restrict__ agg, const float* __restrict__ b,
                          long total, int fmask) {
    long stride = (long)gridDim.x * blockDim.x;
    for (long i = (long)blockIdx.x * blockDim.x + threadIdx.x; i < total; i += stride)
        agg[i] = b[i & fmask];
}

// ---------------------------------------------------------------------------
// COO scatter: agg[dst[e]] += m[src[e]]  (per-feature f32 atomics, float4 gather)
// fshift = log2(F); each thread handles 4 consecutive features of one edge.
// ---------------------------------------------------------------------------
__global__ void scatter_add4(const float* __restrict__ m,
                             const int* __restrict__ src,
                             const int* __restrict__ dst,
                             float* __restrict__ agg,
                             long total4, int fshift) {
    const int  fq    = fshift - 2;
    const long fqm   = (1L << fq) - 1;
    long stride = (long)gridDim.x * blockDim.x;
    for (long idx = (long)blockIdx.x * blockDim.x + threadIdx.x; idx < total4; idx += stride) {
        long e = idx >> fq;
        int  f = (int)((idx & fqm) << 2);
        const float4 v = *(const float4*)(m + ((size_t)src[e] << fshift) + f);
        float* o = agg + ((size_t)dst[e] << fshift) + f;
        atomicAdd(o + 0, v.x);
        atomicAdd(o + 1, v.y);
        atomicAdd(o + 2, v.z);
        atomicAdd(o + 3, v.w);
    }
}

// ---------------------------------------------------------------------------
// out[i] = max(in[i], 0)
// ---------------------------------------------------------------------------
__global__ void relu_copy(const float* __restrict__ in, float* __restrict__ out,
                          long total) {
    long stride = (long)gridDim.x * blockDim.x;
    for (long i = (long)blockIdx.x * blockDim.x + threadIdx.x; i < total; i += stride)
        out[i] = fmaxf(in[i], 0.0f);
}

// ---------------------------------------------------------------------------
extern "C" void kernel_launch(void* const* d_in, const int* in_sizes, int n_in,
                              void* d_out, int out_size, void* d_ws, size_t ws_size,
                              hipStream_t stream) {
    const float* h    = (const float*)d_in[0];
    const int*   edge = (const int*)  d_in[1];
    const float* W1   = (const float*)d_in[2];
    const float* b1   = (const float*)d_in[3];
    const float* W2   = (const float*)d_in[4];
    const float* b2   = (const float*)d_in[5];
    const float* Wa1  = (const float*)d_in[6];
    const float* ba1  = (const float*)d_in[7];
    const float* Wa2  = (const float*)d_in[8];
    const float* ba2  = (const float*)d_in[9];
    const float* Ws   = (const float*)d_in[10];
    const float* bs   = (const float*)d_in[11];

    const int E = in_sizes[1] / 2;
    const int* src = edge;
    const int* dst = edge + E;

    // workspace layout (floats): mtmp[N*512] | agg[N*512] | bufA[N*64] | bufB[N*64]
    float* ws   = (float*)d_ws;
    float* mtmp = ws;
    float* agg  = ws + (size_t)NNODES * FDIM;
    float* bufA = agg + (size_t)NNODES * FDIM;
    float* bufB = bufA + (size_t)NNODES * HDIM;

    float* out        = (float*)d_out;
    float* struct_out = out;                               // [N, N]
    float* xhat_out   = out + (size_t)NNODES * NNODES;     // [N, FDIM]

    auto gemm = [&](const float* A, const float* B, float* C, int K, int Fout) {
        int tilesX = Fout / 16;
        int total  = (NNODES / 16) * tilesX;
        int blocks = (total + 7) / 8;                      // 8 waves / 256-thread block
        gemm_wmma_f32<<<blocks, 256, 0, stream>>>(A, B, C, K, Fout, total, tilesX);
    };

    auto ew_blocks = [](long total) {
        long b = (total + 255) / 256;
        return (int)(b < 16384 ? b : 16384);
    };

    // GCN layer: out = relu( segment_sum((x@W)[src] -> dst) + bias )
    auto gcn = [&](const float* xin, const float* W, const float* bias,
                   float* layer_out, int K, int Fout) {
        gemm(xin, W, mtmp, K, Fout);
        long total = (long)NNODES * Fout;
        bias_init<<<ew_blocks(total), 256, 0, stream>>>(agg, bias, total, Fout - 1);
        int  fshift = (Fout == FDIM) ? 9 : 6;
        long total4 = (long)E * (Fout / 4);
        scatter_add4<<<ew_blocks(total4), 256, 0, stream>>>(mtmp, src, dst, agg,
                                                            total4, fshift);
        relu_copy<<<ew_blocks(total), 256, 0, stream>>>(agg, layer_out, total);
    };

    // shared encoder
    gcn(h,    W1,  b1,  bufA, FDIM, HDIM);   // x1
    gcn(bufA, W2,  b2,  bufB, HDIM, HDIM);   // x2 (kept: feeds both decoders)
    // attribute decoder
    gcn(bufB, Wa1, ba1, bufA, HDIM, HDIM);   // xa (x1 dead, reuse bufA)
    gcn(bufA, Wa2, ba2, xhat_out, HDIM, FDIM); // x_hat -> d_out tail
    // structure decoder
    gcn(bufB, Ws,  bs,  bufA, HDIM, HDIM);   // s  (xa dead, reuse bufA)

    dim3 g(NNODES / 16 / 8, NNODES / 16);    // (96, 768), 8 waves per block
    struct_wmma<<<g, 256, 0, stream>>>(bufA, struct_out);
}